// AbsolutePositionEncoding_40252433498315
// MI455X (gfx1250) — compile-verified
//
#include <hip/hip_runtime.h>
#include <hip/hip_bf16.h>

// ---------------------------------------------------------------------------
// AbsolutePositionEncoding: out[b, s, :] = E[s/8, :]
//   b in [0,64), s in [0,2048), E is [512,256] f32.  Output = 128 MiB f32.
// Pure broadcast: 0 FLOPs, store-bandwidth bound (~5.8 us at 23.3 TB/s).
// Strategy:
//   - grid (512 objects, 8 copy-slices), 256 threads (8 wave32) per block.
//   - Wave 0 stages the 1 KB E row into LDS via the CDNA5 Tensor Data Mover
//     (tensor_load_to_lds + s_wait_tensorcnt), then all waves broadcast it
//     out with fully-coalesced global_store_b128 (512 B per wave per store).
// ---------------------------------------------------------------------------

#define BATCH   64
#define S_LEN   2048
#define EDIM    256          // floats per row (1 KB)
#define OBJS    512
#define COPIES  (BATCH * 8)  // 512 row-copies per object
#define SLICES  8            // blocks per object; each block does 64 copies

typedef unsigned int uint32x4 __attribute__((ext_vector_type(4)));
typedef int          int32x8  __attribute__((ext_vector_type(8)));
typedef int          int32x4  __attribute__((ext_vector_type(4)));

__global__ void __launch_bounds__(256)
ape_broadcast_kernel(const float* __restrict__ E, float* __restrict__ out) {
    const int obj = blockIdx.x;            // object index 0..511
    __shared__ __align__(16) float row[EDIM];   // 1 KB staging tile

#if __has_builtin(__builtin_amdgcn_tensor_load_to_lds) && __has_builtin(__builtin_amdgcn_s_wait_tensorcnt)
    // --- Stage E[obj, 0:256] (1 KB) into LDS with the Tensor Data Mover. ---
    if ((threadIdx.x >> 5) == 0) {         // wave 0 only issues the TDM op
        unsigned long long ga =
            (unsigned long long)(const void*)(E + (size_t)obj * EDIM);
        // LDS flat addresses carry the raw LDS offset in the low 32 bits.
        unsigned lds_base = (unsigned)(unsigned long long)(void*)row;

        // D# group 0: count=1 (valid, user mode), lds_addr, global_addr, type=2.
        uint32x4 g0;
        g0.x = 1u;                                           // count=1
        g0.y = lds_base;                                     // lds_addr[31:0]
        g0.z = (unsigned)(ga & 0xFFFFFFFFu);                 // global_addr[31:0]
        g0.w = (unsigned)((ga >> 32) & 0x01FFFFFFu)          // global_addr[56:32]
             | (2u << 30);                                   // type=2 ("image")

        // D# group 1: 1-D tile of 256 x 4-byte elements.
        int32x8 g1;
        g1[0] = (int)(2u << 16);        // workgroup_mask=0, data_size=2 (4 B)
        g1[1] = (int)((unsigned)EDIM << 16); // tensor_dim0[15:0] = 256
        g1[2] = (int)(1u << 16);        // tensor_dim0[31:16]=0, tensor_dim1=1
        g1[3] = (int)((unsigned)EDIM << 16); // tensor_dim1 hi=0, tile_dim0=256
        g1[4] = 1;                      // tile_dim1=1, tile_dim2=0
        g1[5] = EDIM;                   // tensor_dim0_stride = 256 elements
        g1[6] = 0;                      // stride hi / tensor_dim1_stride lo
        g1[7] = 0;                      // tensor_dim1_stride hi
        int32x4 gz4 = {0, 0, 0, 0};     // groups 2/3 unused (<= 2-D tensor)
        int32x8 gz8 = {0, 0, 0, 0, 0, 0, 0, 0}; // group 4 unused

        // 6-arg toolchain signature: (g0, g1, g2, g3, g4, cpol)
        __builtin_amdgcn_tensor_load_to_lds(g0, g1, gz4, gz4, gz8, 0 /*cpol*/);
        __builtin_amdgcn_s_wait_tensorcnt(0);
    }
#else
    // Fallback staging: 64 lanes x b128 loads.
    if (threadIdx.x < 64) {
        ((float4*)row)[threadIdx.x] =
            ((const float4*)(E + (size_t)obj * EDIM))[threadIdx.x];
    }
#endif
    __syncthreads();

    // --- Broadcast: each thread owns one float4 of the row. ---
    // Lanes of a wave cover 32 consecutive float4 -> contiguous 512 B stores.
    const int q   = threadIdx.x & 63;   // float4 slot within row (0..63)
    const int sub = threadIdx.x >> 6;   // which of 4 concurrent copies (0..3)
    const float4 v = ((const float4*)row)[q];   // ds_load_b128

    float4* __restrict__ o = (float4*)out;
    const int cbase = blockIdx.y * (COPIES / SLICES);   // 64 copies per block

#pragma unroll
    for (int k = 0; k < (COPIES / SLICES) / 4; ++k) {   // 16 b128 stores/thread
        const int c = cbase + sub + (k << 2);           // copy id 0..511
        const int b = c >> 3;                           // batch   0..63
        const int j = c & 7;                            // s within object
        // row index r = b*2048 + obj*8 + j ; float4 index = r*64 + q
        const size_t r = ((size_t)b << 11) + ((size_t)obj << 3) + (size_t)j;
        o[(r << 6) + (size_t)q] = v;                    // global_store_b128
    }
}

extern "C" void kernel_launch(void* const* d_in, const int* in_sizes, int n_in,
                              void* d_out, int out_size, void* d_ws, size_t ws_size,
                              hipStream_t stream) {
    // d_in[0] = x [64,2048] f32 (only its shape matters -> unused)
    // d_in[1] = E_absolute_position [512,256] f32
    const float* E = (const float*)d_in[1];
    float* out = (float*)d_out;     // [64, 2048, 256] f32

    dim3 grid(OBJS, SLICES);        // 4096 blocks
    dim3 block(256);                // 8 wave32 per block
    ape_broadcast_kernel<<<grid, block, 0, stream>>>(E, out);
}